// CayleySolitonPropagator_25142738551437
// MI455X (gfx1250) — compile-verified
//
#include <hip/hip_runtime.h>
#include <math.h>

// CDNA5 / gfx1250: wave32, WMMA f32 16x16x4 for the circulant banded matvec.
// 1 workgroup = 16 rows (WMMA N), 1024 threads = 32 waves, 2 d-tiles per wave.
// LDS: p (neighbor access), x (owner-cell accumulator). Registers: r, Ap.

typedef float v2f __attribute__((ext_vector_type(2)));
typedef float v8f __attribute__((ext_vector_type(8)));

#define DIMD      1024
#define DP        1028      // padded LDS row stride (DP % 64 == 4 -> conflict-free B loads)
#define ROWS_WG   16
#define THREADS   1024
#define TPW       2         // d-tiles (16 wide) per wave
#define HDT       0.05f     // dt/2
#define CG_ITERS  20

// Circulant band matvec for one 16(d) x 16(row) tile, both components.
// pr/pi are row-base pointers (s_p? + m*DP); ctabb = s_ctab + (15 + k0 - m).
// dbs is a wave-uniform (readfirstlane'd) tile base -> scalar branch, full EXEC
// on both sides, and immediate ds offsets on the interior fast path.
__device__ __forceinline__ void hmv_tile(const float* __restrict__ pr,
                                         const float* __restrict__ pi,
                                         const float* __restrict__ ctabb,
                                         int dbs, int k0,
                                         v8f& accR, v8f& accI)
{
    if (dbs >= 32 && dbs <= DIMD - 48) {
        // interior: window [dbs-32, dbs+48) needs no wrap -> one base, imm offsets
        const float* bp_r = pr + (dbs - 32 + k0);
        const float* bp_i = pi + (dbs - 32 + k0);
#pragma unroll
        for (int c = 0; c < 20; ++c) {
            v2f a;
            a[0] = ctabb[4 * c];
            a[1] = ctabb[4 * c + 1];
            const v2f br = *(const v2f*)(bp_r + 4 * c);
            const v2f bi = *(const v2f*)(bp_i + 4 * c);
            accR = __builtin_amdgcn_wmma_f32_16x16x4_f32(false, a, false, br, (short)0, accR, false, false);
            accI = __builtin_amdgcn_wmma_f32_16x16x4_f32(false, a, false, bi, (short)0, accI, false, false);
        }
    } else {
        // boundary tiles (4 of 64): circulant wrap per 4-chunk
#pragma unroll
        for (int c = 0; c < 20; ++c) {
            const int rel = 4 * c - 32;
            const int dd  = (dbs + rel + DIMD) & (DIMD - 1);
            v2f a;
            a[0] = ctabb[4 * c];
            a[1] = ctabb[4 * c + 1];
            const v2f br = *(const v2f*)(pr + dd + k0);
            const v2f bi = *(const v2f*)(pi + dd + k0);
            accR = __builtin_amdgcn_wmma_f32_16x16x4_f32(false, a, false, br, (short)0, accR, false, false);
            accI = __builtin_amdgcn_wmma_f32_16x16x4_f32(false, a, false, bi, (short)0, accI, false, false);
        }
    }
}

__global__ __launch_bounds__(THREADS, 1)
void cayley_cg_kernel(const float* __restrict__ g_psi_r,
                      const float* __restrict__ g_psi_i,
                      const float* __restrict__ g_alpha,
                      const float* __restrict__ g_sw,
                      const float* __restrict__ g_pot,
                      float* __restrict__ g_out)
{
    __shared__ __align__(16) float s_pr[ROWS_WG * DP];   // p real (psi/rot staging first)
    __shared__ __align__(16) float s_pi[ROWS_WG * DP];   // p imag
    __shared__ __align__(16) float s_xr[ROWS_WG * DP];   // x real accumulator
    __shared__ __align__(16) float s_xi[ROWS_WG * DP];   // x imag accumulator
    __shared__ float s_alpha[DIMD];
    __shared__ float s_pot[DIMD];
    __shared__ float s_ctab[96];
    __shared__ float s_red[1024];
    __shared__ float s_rs[16];
    __shared__ float s_al[16];
    __shared__ float s_bt[16];
    __shared__ float s_f[16];

    const int tid  = threadIdx.x;
    const int lane = tid & 31;
    const int wave = tid >> 5;
    const int m    = lane & 15;      // row (N for B/C/D, M for A)
    const int half = lane >> 4;
    const int k0   = half * 2;
    const int R0   = blockIdx.x * ROWS_WG;

    const float* prow_r = s_pr + m * DP;
    const float* prow_i = s_pi + m * DP;
    const float* ctabb  = s_ctab + (15 + k0 - m);  // delta = -32 + 4c + k - m  ->  +47

    // ---------- phase 0: load psi -> LDS, zero x, tables, intensity partials ----------
    const int lr = tid >> 6;
    const int lc = tid & 63;
    float ipart = 0.f;
#pragma unroll
    for (int j = 0; j < 16; ++j) {
        const int d = lc + j * 64;
        const float vr = g_psi_r[(size_t)(R0 + lr) * DIMD + d];
        const float vi = g_psi_i[(size_t)(R0 + lr) * DIMD + d];
        s_pr[lr * DP + d] = vr;
        s_pi[lr * DP + d] = vi;
        s_xr[lr * DP + d] = 0.f;
        s_xi[lr * DP + d] = 0.f;
        ipart += vr * vr + vi * vi;
    }
    s_red[lr * 64 + lc] = ipart;
    s_alpha[tid] = g_alpha[tid];
    s_pot[tid]   = g_pot[tid];
    if (tid < 96) {
        const float w0 = g_sw[0], w1 = g_sw[1], w2 = g_sw[2];
        int ad = tid - 47; if (ad < 0) ad = -ad;
        float v = 0.f;
        if      (ad == 0)                            v = 10.f * (w0 + w1 + w2);
        else if (ad == 1 || ad == 3 || ad == 5)      v = -w0;
        else if (ad == 2)                            v = -(w0 + w1);
        else if (ad == 4)                            v = -(w0 + w1 + w2);
        else if (ad == 6 || ad == 10)                v = -w1;
        else if (ad == 8)                            v = -(w1 + w2);
        else if (ad == 12 || ad == 16 || ad == 20)   v = -w2;
        s_ctab[tid] = v;
    }
    __syncthreads();

    if (tid < 16) {
        float s = 0.f;
#pragma unroll
        for (int k = 0; k < 64; ++k) s += s_red[tid * 64 + k];
        s_f[tid] = 1.f / (s * (1.f / (float)DIMD) + 1e-8f);
    }
    __syncthreads();

    // ---------- phase 1: nonlinear phase rotation (in place in LDS) ----------
    {
        const float f = s_f[lr];
#pragma unroll
        for (int j = 0; j < 16; ++j) {
            const int d = lc + j * 64;
            const float vr = s_pr[lr * DP + d];
            const float vi = s_pi[lr * DP + d];
            const float ph = s_alpha[d] * ((vr * vr + vi * vi) * f);
            const float c = cosf(ph), sn = sinf(ph);
            s_pr[lr * DP + d] = vr * c - vi * sn;
            s_pi[lr * DP + d] = vr * sn + vi * c;
        }
    }
    __syncthreads();

    // ---------- phase 2: rhs = (I - i*hdt*H) rot ; r = p = rhs ----------
    float rreg[TPW][2][8];
    float apc [TPW][2][8];
    float part = 0.f;
#pragma unroll
    for (int t = 0; t < TPW; ++t) {
        const int dbs = __builtin_amdgcn_readfirstlane((wave * TPW + t) * 16);
        v8f aR = (v8f)0.0f, aI = (v8f)0.0f;
        hmv_tile(prow_r, prow_i, ctabb, dbs, k0, aR, aI);
        const int base = m * DP + dbs + half * 8;
        const int pb   = dbs + half * 8;
#pragma unroll
        for (int g = 0; g < 8; ++g) {
            const float rr = s_pr[base + g];
            const float ri = s_pi[base + g];
            const float po = s_pot[pb + g];
            const float hr = aR[g] + po * rr;
            const float hi = aI[g] + po * ri;
            const float br = rr + HDT * hi;
            const float bi = ri - HDT * hr;
            rreg[t][0][g] = br;
            rreg[t][1][g] = bi;
            part += br * br + bi * bi;
        }
    }
    part += __shfl_xor(part, 16, 32);
    if (half == 0) s_red[wave * 17 + m] = part;
    __syncthreads();
    if (tid < 16) {
        float s = 0.f;
#pragma unroll
        for (int w = 0; w < 32; ++w) s += s_red[w * 17 + tid];
        s_rs[tid] = s;
    }
    __syncthreads();                 // fences Hmv reads of rot before p overwrite
#pragma unroll
    for (int t = 0; t < TPW; ++t) {  // p = rhs
        const int dbs = __builtin_amdgcn_readfirstlane((wave * TPW + t) * 16);
        const int base = m * DP + dbs + half * 8;
#pragma unroll
        for (int g = 0; g < 8; ++g) {
            s_pr[base + g] = rreg[t][0][g];
            s_pi[base + g] = rreg[t][1][g];
        }
    }

    // ---------- phase 3: CG (fixed 20 iterations) ----------
    for (int it = 0; it < CG_ITERS; ++it) {
        __syncthreads();             // p stores visible to all Hmv readers
        float pAp = 0.f;
#pragma unroll
        for (int t = 0; t < TPW; ++t) {
            const int dbs = __builtin_amdgcn_readfirstlane((wave * TPW + t) * 16);
            v8f aR = (v8f)0.0f, aI = (v8f)0.0f;
            hmv_tile(prow_r, prow_i, ctabb, dbs, k0, aR, aI);
            const int base = m * DP + dbs + half * 8;
            const int pb   = dbs + half * 8;
#pragma unroll
            for (int g = 0; g < 8; ++g) {
                const float pr_ = s_pr[base + g];
                const float pi_ = s_pi[base + g];
                const float po  = s_pot[pb + g];
                const float hr  = aR[g] + po * pr_;
                const float hi  = aI[g] + po * pi_;
                const float apr = pr_ - HDT * hi;   // (I + i*hdt*H) in real-pair form
                const float api = pi_ + HDT * hr;
                apc[t][0][g] = apr;
                apc[t][1][g] = api;
                pAp += pr_ * apr + pi_ * api;
            }
        }
        pAp += __shfl_xor(pAp, 16, 32);
        if (half == 0) s_red[wave * 17 + m] = pAp;
        __syncthreads();
        if (tid < 16) {
            float s = 0.f;
#pragma unroll
            for (int w = 0; w < 32; ++w) s += s_red[w * 17 + tid];
            s_al[tid] = s_rs[tid] / (s + 1e-12f);
        }
        __syncthreads();
        const float al = s_al[m];
        float rsn = 0.f;
#pragma unroll
        for (int t = 0; t < TPW; ++t) {
            const int dbs = __builtin_amdgcn_readfirstlane((wave * TPW + t) * 16);
            const int base = m * DP + dbs + half * 8;
#pragma unroll
            for (int g = 0; g < 8; ++g) {
                const float pr_ = s_pr[base + g];   // p still valid in LDS
                const float pi_ = s_pi[base + g];
                s_xr[base + g] += al * pr_;         // owner-cell RMW
                s_xi[base + g] += al * pi_;
                const float nr = rreg[t][0][g] - al * apc[t][0][g];
                const float ni = rreg[t][1][g] - al * apc[t][1][g];
                rreg[t][0][g] = nr;
                rreg[t][1][g] = ni;
                rsn += nr * nr + ni * ni;
            }
        }
        rsn += __shfl_xor(rsn, 16, 32);
        if (half == 0) s_red[wave * 17 + m] = rsn;
        __syncthreads();
        if (tid < 16) {
            float s = 0.f;
#pragma unroll
            for (int w = 0; w < 32; ++w) s += s_red[w * 17 + tid];
            s_bt[tid] = s / (s_rs[tid] + 1e-12f);
            s_rs[tid] = s;
        }
        __syncthreads();
        const float bt = s_bt[m];
#pragma unroll
        for (int t = 0; t < TPW; ++t) {   // p = r + beta*p (owner-cell RMW)
            const int dbs = __builtin_amdgcn_readfirstlane((wave * TPW + t) * 16);
            const int base = m * DP + dbs + half * 8;
#pragma unroll
            for (int g = 0; g < 8; ++g) {
                s_pr[base + g] = rreg[t][0][g] + bt * s_pr[base + g];
                s_pi[base + g] = rreg[t][1][g] + bt * s_pi[base + g];
            }
        }
    }

    // ---------- phase 4: coalesced interleaved store of x ----------
    __syncthreads();
    float2* outv = (float2*)g_out;        // output is [row][d][2] interleaved
#pragma unroll
    for (int j = 0; j < 16; ++j) {
        const int d = lc + j * 64;
        float2 o;
        o.x = s_xr[lr * DP + d];
        o.y = s_xi[lr * DP + d];
        outv[(size_t)(R0 + lr) * DIMD + d] = o;
    }
}

extern "C" void kernel_launch(void* const* d_in, const int* in_sizes, int n_in,
                              void* d_out, int out_size, void* d_ws, size_t ws_size,
                              hipStream_t stream) {
    (void)n_in; (void)out_size; (void)d_ws; (void)ws_size;
    const float* psi_r = (const float*)d_in[0];
    const float* psi_i = (const float*)d_in[1];
    const float* alpha = (const float*)d_in[2];
    const float* sw    = (const float*)d_in[3];
    const float* pot   = (const float*)d_in[4];
    float* out = (float*)d_out;

    const int rows = in_sizes[0] / DIMD;       // B*S = 8192
    const int grid = rows / ROWS_WG;           // 512 workgroups, 16 rows each
    cayley_cg_kernel<<<grid, THREADS, 0, stream>>>(psi_r, psi_i, alpha, sw, pot, out);
}